// Net_47433618817573
// MI455X (gfx1250) — compile-verified
//
#include <hip/hip_runtime.h>
#include <math.h>

typedef float v2f __attribute__((ext_vector_type(2)));
typedef float v8f __attribute__((ext_vector_type(8)));

namespace cfg {
constexpr int IMG = 1024;
constexpr int ZD  = 128;
constexpr int AD  = 6;
constexpr int HH  = 128;
constexpr int NT  = 4;    // timesteps
constexpr int NB  = 64;   // batch
constexpr float OFFC = 3.0f;
constexpr long long PE_T = 148480;
constexpr long long DI_T = 148608;
constexpr long long DP_T = 17286;
constexpr long long RN_T = 32896;
// offsets inside generated per-batch parameter vectors
constexpr long long PE_W1 = 0,     PE_B1 = 131840, PE_W2 = 131968, PE_B2 = 148352;
constexpr long long DI_W1 = 0,     DI_B1 = 16384,  DI_W2 = 16512,  DI_B2 = 147584;
constexpr long long DP_W1 = 0,     DP_B1 = 16384,  DP_W2 = 16512,  DP_B2 = 17280;
constexpr long long RN_WI = 0,     RN_WH = 16384,  RN_B  = 32768;
// output packing
constexpr int OUT_OFF1 = NB*NT*ZD;            // 32768  (z_ts)
constexpr int OUT_OFF2 = OUT_OFF1 + NB*NT*AD; // 34304  (a_ts)
constexpr int OUT_OFF3 = OUT_OFF2 + NB*NT*IMG;// 296448 (x_ts)
constexpr int OUT_OFF4 = OUT_OFF3 + NB*NT*IMG;// 558592 (patches)
constexpr int OUT_TOT  = OUT_OFF4 + NB*NT*IMG;// 820736 (origs end)
}
using namespace cfg;

enum { ACT_NONE = 0, ACT_RELU = 1, ACT_SIG = 2, ACT_SIN = 3, ACT_TANH = 4 };

__device__ __forceinline__ float applyAct(float v, int act) {
  switch (act) {
    case ACT_RELU: return v > 0.f ? v : 0.f;
    case ACT_SIG:  return 1.f / (1.f + expf(-v));
    case ACT_SIN:  return sinf(v);
    case ACT_TANH: return tanhf(v);
    default:       return v;
  }
}
__device__ __forceinline__ int clampi(int v, int lo, int hi) {
  return v < lo ? lo : (v > hi ? hi : v);
}

// ---------------------------------------------------------------------------
// Generic WMMA f32 GEMM: C[M,N] = act(A[M,K] @ B[K,N] + bias[N])
// One wave per 16x32 C slab (two 16x16 WMMA tiles share one A fragment).
// Main K loop is branchless: out-of-range rows/cols are clamped to a valid
// address; their products land in C rows/cols that are never stored.
// K tail (K%4 != 0) peeled and guarded. Requires K even (all call sites).
// ---------------------------------------------------------------------------
__global__ void nk_gemm(const float* __restrict__ A, const float* __restrict__ Bm,
                        const float* __restrict__ bias, float* __restrict__ C,
                        int M, int N, int K, int act) {
  const int lane = threadIdx.x;
  const int m0 = blockIdx.y * 16;
  const int n0 = blockIdx.x * 32;
  const int mrow = m0 + (lane & 15);
  const int nc0 = n0 + (lane & 15);
  const int nc1 = nc0 + 16;
  const int kh = (lane >> 4) * 2;
  const int msafe = mrow < M ? mrow : M - 1;
  const int ns0 = nc0 < N ? nc0 : N - 1;
  const int ns1 = nc1 < N ? nc1 : N - 1;
  const float* Ap = A + (long long)msafe * K + kh;
  const float* B0 = Bm + (long long)kh * N + ns0;
  const float* B1 = Bm + (long long)kh * N + ns1;
  const long long stepB = (long long)4 * N;
  v8f acc0 = {0.f, 0.f, 0.f, 0.f, 0.f, 0.f, 0.f, 0.f};
  v8f acc1 = {0.f, 0.f, 0.f, 0.f, 0.f, 0.f, 0.f, 0.f};
  int k = 0;
  for (; k + 4 <= K; k += 4) {
    const v2f av = *(const v2f*)Ap;              // global_load_b64
    v2f b0, b1;
    b0.x = B0[0]; b0.y = B0[N];
    b1.x = B1[0]; b1.y = B1[N];
    acc0 = __builtin_amdgcn_wmma_f32_16x16x4_f32(false, av, false, b0, (short)0, acc0,
                                                 false, false);
    acc1 = __builtin_amdgcn_wmma_f32_16x16x4_f32(false, av, false, b1, (short)0, acc1,
                                                 false, false);
    Ap += 4; B0 += stepB; B1 += stepB;
  }
  if (k < K) {  // guarded K tail
    const int ka = k + kh;
    v2f av, b0, b1;
    av.x = (ka < K)     ? A[(long long)msafe * K + ka]     : 0.f;
    av.y = (ka + 1 < K) ? A[(long long)msafe * K + ka + 1] : 0.f;
    b0.x = (ka < K)     ? Bm[(long long)ka * N + ns0]       : 0.f;
    b0.y = (ka + 1 < K) ? Bm[(long long)(ka + 1) * N + ns0] : 0.f;
    b1.x = (ka < K)     ? Bm[(long long)ka * N + ns1]       : 0.f;
    b1.y = (ka + 1 < K) ? Bm[(long long)(ka + 1) * N + ns1] : 0.f;
    acc0 = __builtin_amdgcn_wmma_f32_16x16x4_f32(false, av, false, b0, (short)0, acc0,
                                                 false, false);
    acc1 = __builtin_amdgcn_wmma_f32_16x16x4_f32(false, av, false, b1, (short)0, acc1,
                                                 false, false);
  }
  const int mbase = m0 + ((lane >> 4) ? 8 : 0);
  if (nc0 < N) {
    const float bb = bias ? bias[nc0] : 0.f;
#pragma unroll
    for (int v = 0; v < 8; ++v) {
      const int cm = mbase + v;
      if (cm < M) C[(long long)cm * N + nc0] = applyAct(acc0[v] + bb, act);
    }
  }
  if (nc1 < N) {
    const float bb = bias ? bias[nc1] : 0.f;
#pragma unroll
    for (int v = 0; v < 8; ++v) {
      const int cm = mbase + v;
      if (cm < M) C[(long long)cm * N + nc1] = applyAct(acc1[v] + bb, act);
    }
  }
}

// ---------------------------------------------------------------------------
// WMMA GEMM with transposed B: Cpart[kz][M,N] = A[M,K] @ G[N,K]^T over K-chunks
// (deterministic split-K; partials reduced by nk_reduce). Both fragments are
// K-contiguous -> b64 loads; branchless main loop, guarded K tail.
// ---------------------------------------------------------------------------
__global__ void nk_gemm_bt(const float* __restrict__ A, const float* __restrict__ G,
                           float* __restrict__ Cpart, int M, int N, int K, int kchunk) {
  const int lane = threadIdx.x;
  const int m0 = blockIdx.y * 16;
  const int n0 = blockIdx.x * 16;
  const int kz = blockIdx.z;
  const long long k0 = (long long)kz * kchunk;   // multiple of 4
  long long kend = k0 + kchunk;
  if (kend > K) kend = K;
  const int mrow = m0 + (lane & 15);
  const int ncol = n0 + (lane & 15);
  const int kh = (lane >> 4) * 2;
  const int msafe = mrow < M ? mrow : M - 1;
  const int nsafe = ncol < N ? ncol : N - 1;
  const float* Ap = A + (long long)msafe * K + k0 + kh;
  const float* Gp = G + (long long)nsafe * K + k0 + kh;
  v8f acc = {0.f, 0.f, 0.f, 0.f, 0.f, 0.f, 0.f, 0.f};
  long long k = k0;
  for (; k + 4 <= kend; k += 4) {
    const v2f av = *(const v2f*)Ap;   // global_load_b64
    const v2f bv = *(const v2f*)Gp;   // global_load_b64
    acc = __builtin_amdgcn_wmma_f32_16x16x4_f32(false, av, false, bv, (short)0, acc,
                                                false, false);
    Ap += 4; Gp += 4;
  }
  if (k < kend) {  // guarded K tail
    const long long ka = k + kh;
    v2f av, bv;
    av.x = (ka < kend)     ? A[(long long)msafe * K + ka]     : 0.f;
    av.y = (ka + 1 < kend) ? A[(long long)msafe * K + ka + 1] : 0.f;
    bv.x = (ka < kend)     ? G[(long long)nsafe * K + ka]     : 0.f;
    bv.y = (ka + 1 < kend) ? G[(long long)nsafe * K + ka + 1] : 0.f;
    acc = __builtin_amdgcn_wmma_f32_16x16x4_f32(false, av, false, bv, (short)0, acc,
                                                false, false);
  }
  const int mbase = m0 + ((lane >> 4) ? 8 : 0);
  if (ncol < N) {
#pragma unroll
    for (int v = 0; v < 8; ++v) {
      const int cm = mbase + v;
      if (cm < M)
        Cpart[(long long)kz * M * N + (long long)cm * N + ncol] = acc[v];
    }
  }
}

__global__ void nk_reduce(const float* __restrict__ part, int nz, int len,
                          float* __restrict__ out, int acc) {
  int i = blockIdx.x * blockDim.x + threadIdx.x;
  if (i >= len) return;
  float s = acc ? out[i] : 0.f;
  for (int z = 0; z < nz; ++z) s += part[(long long)z * len + i];
  out[i] = s;
}

// ---------------------------------------------------------------------------
// Batched GEMV with per-batch generated weights: Y[b,o] = act((acc?Y:0)+bias+X[b]·W(b)[:,o])
// ---------------------------------------------------------------------------
__global__ void nk_bgemv(const float* __restrict__ X, int xs,
                         const float* __restrict__ P, long long pstr,
                         long long woff, long long boff,
                         float* __restrict__ Y, int ys,
                         int din, int dout, int accum, int act) {
  const int b = blockIdx.y;
  const int o = blockIdx.x * blockDim.x + threadIdx.x;
  if (o >= dout) return;
  const float* W = P + (long long)b * pstr + woff;
  const float* x = X + (long long)b * xs;
  float s = 0.f;
  for (int i = 0; i < din; ++i) s = fmaf(x[i], W[(long long)i * dout + o], s);
  if (boff >= 0) s += P[(long long)b * pstr + boff + o];
  if (accum) s += Y[(long long)b * ys + o];
  Y[(long long)b * ys + o] = applyAct(s, act);
}

// DX[b,i] = (acc?DX:0) + sum_o DY[b,o] * W(b)[i,o]
__global__ void nk_bgemvT(const float* __restrict__ DY, int ys,
                          const float* __restrict__ P, long long pstr, long long woff,
                          float* __restrict__ DX, int xs,
                          int din, int dout, int accum) {
  const int b = blockIdx.y;
  const int i = blockIdx.x * blockDim.x + threadIdx.x;
  if (i >= din) return;
  const float* W = P + (long long)b * pstr + woff + (long long)i * dout;
  const float* dy = DY + (long long)b * ys;
  float s = accum ? DX[(long long)b * xs + i] : 0.f;
  for (int o = 0; o < dout; ++o) s = fmaf(dy[o], W[o], s);
  DX[(long long)b * xs + i] = s;
}

// dP(b)[woff + i*dout + o] += X[b,i]*DY[b,o]
__global__ void nk_bouter(const float* __restrict__ X, int xs,
                          const float* __restrict__ DY, int ys,
                          float* __restrict__ dP, long long pstr, long long woff,
                          int din, int dout) {
  const int b = blockIdx.y;
  const long long idx = (long long)blockIdx.x * blockDim.x + threadIdx.x;
  if (idx >= (long long)din * dout) return;
  const int i = (int)(idx / dout), o = (int)(idx % dout);
  dP[(long long)b * pstr + woff + idx] +=
      X[(long long)b * xs + i] * DY[(long long)b * ys + o];
}

__global__ void nk_baddb(const float* __restrict__ DY, int ys,
                         float* __restrict__ dP, long long pstr, long long boff,
                         int dout) {
  const int b = blockIdx.y;
  const int o = blockIdx.x * blockDim.x + threadIdx.x;
  if (o >= dout) return;
  dP[(long long)b * pstr + boff + o] += DY[(long long)b * ys + o];
}

// ---------------------------------------------------------------------------
// elementwise
// ---------------------------------------------------------------------------
__global__ void nk_ew_sin(float* dst, const float* src, int n) {
  int i = blockIdx.x * blockDim.x + threadIdx.x;
  if (i < n) dst[i] = sinf(src[i]);
}
__global__ void nk_ew_neg(float* dst, const float* src, int n) {
  int i = blockIdx.x * blockDim.x + threadIdx.x;
  if (i < n) dst[i] = -src[i];
}
__global__ void nk_ew_add(float* dst, const float* a, const float* b, int n) {
  int i = blockIdx.x * blockDim.x + threadIdx.x;
  if (i < n) dst[i] = a[i] + b[i];
}
__global__ void nk_ew_sigbwd(float* dst, const float* y, const float* dy, int n) {
  int i = blockIdx.x * blockDim.x + threadIdx.x;
  if (i < n) { float v = y[i]; dst[i] = dy[i] * v * (1.f - v); }
}
__global__ void nk_ew_tanhbwd(float* dst, const float* y, const float* dy, int n) {
  int i = blockIdx.x * blockDim.x + threadIdx.x;
  if (i < n) { float v = y[i]; dst[i] = dy[i] * (1.f - v * v); }
}
__global__ void nk_ew_sinbwd(float* dst, const float* pre, const float* dy, int n) {
  int i = blockIdx.x * blockDim.x + threadIdx.x;
  if (i < n) dst[i] = dy[i] * cosf(pre[i]);
}
__global__ void nk_ew_relubwd(float* dst, const float* y, const float* dy, int n) {
  int i = blockIdx.x * blockDim.x + threadIdx.x;
  if (i < n) dst[i] = (y[i] > 0.f) ? dy[i] : 0.f;
}
__global__ void nk_concat(const float* __restrict__ x, const float* __restrict__ a,
                          float* __restrict__ inp) {
  int i = blockIdx.x * blockDim.x + threadIdx.x;
  const int n = NB * (IMG + AD);
  if (i >= n) return;
  const int b = i / (IMG + AD), j = i % (IMG + AD);
  inp[i] = (j < IMG) ? x[b * IMG + j] : a[b * AD + (j - IMG)];
}
__global__ void nk_split(const float* __restrict__ dinp, float* __restrict__ dx,
                         float* __restrict__ da) {
  int i = blockIdx.x * blockDim.x + threadIdx.x;
  const int n = NB * (IMG + AD);
  if (i >= n) return;
  const int b = i / (IMG + AD), j = i % (IMG + AD);
  const float v = dinp[i];
  if (j < IMG) dx[b * IMG + j] = v; else da[b * AD + (j - IMG)] = v;
}
__global__ void nk_residual(const float* __restrict__ patch, const float* __restrict__ x,
                            float* __restrict__ r) {
  int i = blockIdx.x * blockDim.x + threadIdx.x;
  if (i >= NB * IMG) return;
  const int b = i / IMG, p = i % IMG;
  float s = 0.f;
  for (int t = 0; t < NT; ++t) s += patch[((long long)t * NB + b) * IMG + p];
  r[i] = (2.f / NB) * (s - x[i]);
}

// ---------------------------------------------------------------------------
// bilinear sampler (align_corners=True, border) forward / backward
// ---------------------------------------------------------------------------
__global__ void nk_patch_fwd(const float* __restrict__ XT, const float* __restrict__ AT,
                             float* __restrict__ PATCH) {
  const int b = blockIdx.y;
  const int p = blockIdx.x * blockDim.x + threadIdx.x;
  if (p >= IMG) return;
  const int h = p >> 5, w = p & 31;
  const float* a = AT + b * AD;
  const float t0 = a[0] + OFFC, t1 = a[1], t2 = a[2];
  const float t3 = a[3], t4 = a[4] + OFFC, t5 = a[5];
  const float X = -1.f + 2.f * w / 31.f;
  const float Y = -1.f + 2.f * h / 31.f;
  const float gx = t0 * X + t1 * Y + t2;
  const float gy = t3 * X + t4 * Y + t5;
  const float ix = (gx + 1.f) * 0.5f * 31.f;
  const float iy = (gy + 1.f) * 0.5f * 31.f;
  const float x0f = floorf(ix), y0f = floorf(iy);
  const float wnw = (x0f + 1.f - ix) * (y0f + 1.f - iy);
  const float wne = (ix - x0f) * (y0f + 1.f - iy);
  const float wsw = (x0f + 1.f - ix) * (iy - y0f);
  const float wse = (ix - x0f) * (iy - y0f);
  const int xi0 = clampi((int)x0f, 0, 31), yi0 = clampi((int)y0f, 0, 31);
  const int xi1 = clampi((int)x0f + 1, 0, 31), yi1 = clampi((int)y0f + 1, 0, 31);
  const float* img = XT + (long long)b * IMG;
  PATCH[(long long)b * IMG + p] = wnw * img[yi0 * 32 + xi0] + wne * img[yi0 * 32 + xi1] +
                                  wsw * img[yi1 * 32 + xi0] + wse * img[yi1 * 32 + xi1];
}

__global__ void nk_patch_bwd(const float* __restrict__ R, const float* __restrict__ XT,
                             const float* __restrict__ AT, float* __restrict__ DX,
                             float* __restrict__ DA) {
  __shared__ float sred[6][256];
  const int b = blockIdx.y;
  const int tid = threadIdx.x;
  const int p = blockIdx.x * blockDim.x + tid;  // always < IMG (grid 4 x 256)
  const int h = p >> 5, w = p & 31;
  const float* a = AT + b * AD;
  const float t0 = a[0] + OFFC, t1 = a[1], t2 = a[2];
  const float t3 = a[3], t4 = a[4] + OFFC, t5 = a[5];
  const float X = -1.f + 2.f * w / 31.f;
  const float Y = -1.f + 2.f * h / 31.f;
  const float gx = t0 * X + t1 * Y + t2;
  const float gy = t3 * X + t4 * Y + t5;
  const float ix = (gx + 1.f) * 0.5f * 31.f;
  const float iy = (gy + 1.f) * 0.5f * 31.f;
  const float x0f = floorf(ix), y0f = floorf(iy);
  const float x1d = (x0f + 1.f - ix), x0d = (ix - x0f);
  const float y1d = (y0f + 1.f - iy), y0d = (iy - y0f);
  const int xi0 = clampi((int)x0f, 0, 31), yi0 = clampi((int)y0f, 0, 31);
  const int xi1 = clampi((int)x0f + 1, 0, 31), yi1 = clampi((int)y0f + 1, 0, 31);
  const float* img = XT + (long long)b * IMG;
  const float vnw = img[yi0 * 32 + xi0], vne = img[yi0 * 32 + xi1];
  const float vsw = img[yi1 * 32 + xi0], vse = img[yi1 * 32 + xi1];
  const float r = R[(long long)b * IMG + p];
  float* dximg = DX + (long long)b * IMG;
  atomicAdd(&dximg[yi0 * 32 + xi0], r * x1d * y1d);
  atomicAdd(&dximg[yi0 * 32 + xi1], r * x0d * y1d);
  atomicAdd(&dximg[yi1 * 32 + xi0], r * x1d * y0d);
  atomicAdd(&dximg[yi1 * 32 + xi1], r * x0d * y0d);
  const float dix = r * (-y1d * vnw + y1d * vne - y0d * vsw + y0d * vse);
  const float diy = r * (-x1d * vnw - x0d * vne + x1d * vsw + x0d * vse);
  const float dgx = dix * 15.5f, dgy = diy * 15.5f;
  sred[0][tid] = dgx * X; sred[1][tid] = dgx * Y; sred[2][tid] = dgx;
  sred[3][tid] = dgy * X; sred[4][tid] = dgy * Y; sred[5][tid] = dgy;
  __syncthreads();
  for (int s = 128; s > 0; s >>= 1) {
    if (tid < s)
      for (int c = 0; c < 6; ++c) sred[c][tid] += sred[c][tid + s];
    __syncthreads();
  }
  if (tid == 0)
    for (int c = 0; c < 6; ++c) atomicAdd(&DA[b * AD + c], sred[c][0]);
}

// _zoom_in (align_corners=False, zeros padding), stop-gradient path.
__global__ void nk_zoom(const float* __restrict__ XIN, const float* __restrict__ AT,
                        float* __restrict__ ORIG) {
  const int t = blockIdx.z;
  const int b = blockIdx.y;
  const int p = blockIdx.x * blockDim.x + threadIdx.x;
  if (p >= IMG) return;
  const int h = p >> 5, w = p & 31;
  const float* a = AT + ((long long)t * NB + b) * AD;
  const float tn0 = a[0] + OFFC, tn1 = a[1] + OFFC, tn2 = a[2] + OFFC;
  const float tn3 = a[3] + OFFC, tn4 = a[4] + OFFC, tn5 = a[5] + OFFC;
  const float th00 = 1.f / tn0, th01 = tn1, th02 = tn0 * tn2;
  const float th10 = tn3, th11 = 1.f / tn4, th12 = tn4 * tn5;
  const float X = (2.f * w + 1.f) / 32.f - 1.f;
  const float Y = (2.f * h + 1.f) / 32.f - 1.f;
  const float gx = th00 * X + th01 * Y + th02;
  const float gy = th10 * X + th11 * Y + th12;
  const float ix = ((gx + 1.f) * 32.f - 1.f) * 0.5f;
  const float iy = ((gy + 1.f) * 32.f - 1.f) * 0.5f;
  const float x0f = floorf(ix), y0f = floorf(iy);
  const float x1f = x0f + 1.f, y1f = y0f + 1.f;
  const float* img = XIN + (long long)b * IMG;
  auto corner = [&](float xc, float yc, float wt) -> float {
    const bool in = (xc >= 0.f) && (xc <= 31.f) && (yc >= 0.f) && (yc <= 31.f);
    const int xi = clampi((int)fminf(fmaxf(xc, 0.f), 31.f), 0, 31);
    const int yi = clampi((int)fminf(fmaxf(yc, 0.f), 31.f), 0, 31);
    return in ? img[yi * 32 + xi] * wt : 0.f;
  };
  ORIG[((long long)t * NB + b) * IMG + p] =
      corner(x0f, y0f, (x1f - ix) * (y1f - iy)) + corner(x1f, y0f, (ix - x0f) * (y1f - iy)) +
      corner(x0f, y1f, (x1f - ix) * (iy - y0f)) + corner(x1f, y1f, (ix - x0f) * (iy - y0f));
}

__global__ void nk_pack(const float* __restrict__ hs, const float* __restrict__ at,
                        const float* __restrict__ xt, const float* __restrict__ patch,
                        const float* __restrict__ orig, float* __restrict__ out) {
  int i = blockIdx.x * blockDim.x + threadIdx.x;
  if (i >= OUT_TOT) return;
  if (i < OUT_OFF1) {
    const int b = i / (NT * ZD), r = i % (NT * ZD), t = r / ZD, k = r % ZD;
    out[i] = hs[((long long)t * NB + b) * ZD + k];
  } else if (i < OUT_OFF2) {
    const int j = i - OUT_OFF1;
    const int b = j / (NT * AD), r = j % (NT * AD), t = r / AD, k = r % AD;
    out[i] = at[((long long)t * NB + b) * AD + k];
  } else if (i < OUT_OFF3) {
    const int j = i - OUT_OFF2;
    const int b = j / (NT * IMG), r = j % (NT * IMG), t = r / IMG, k = r % IMG;
    out[i] = xt[((long long)t * NB + b) * IMG + k];
  } else if (i < OUT_OFF4) {
    const int j = i - OUT_OFF3;
    const int b = j / (NT * IMG), r = j % (NT * IMG), t = r / IMG, k = r % IMG;
    out[i] = patch[((long long)t * NB + b) * IMG + k];
  } else {
    const int j = i - OUT_OFF4;
    const int b = j / (NT * IMG), r = j % (NT * IMG), t = r / IMG, k = r % IMG;
    out[i] = orig[((long long)t * NB + b) * IMG + k];
  }
}

// ---------------------------------------------------------------------------
// host orchestration
// ---------------------------------------------------------------------------
extern "C" void kernel_launch(void* const* d_in, const int* in_sizes, int n_in,
                              void* d_out, int out_size, void* d_ws, size_t ws_size,
                              hipStream_t stream) {
  (void)in_sizes; (void)n_in; (void)out_size; (void)ws_size;
  const float* x    = (const float*)d_in[0];
  const float* hzW0 = (const float*)d_in[1];
  const float* hzb0 = (const float*)d_in[2];
  const float* hzW  = (const float*)d_in[3];
  const float* hzb  = (const float*)d_in[4];
  const float* ipW1 = (const float*)d_in[5];
  const float* ipb1 = (const float*)d_in[6];
  const float* ipW2 = (const float*)d_in[7];
  const float* ipb2 = (const float*)d_in[8];
  const float* iolW1 = (const float*)d_in[9];
  const float* iolb1 = (const float*)d_in[10];
  const float* iolW2 = (const float*)d_in[11];
  const float* iolb2 = (const float*)d_in[12];
  const float* peG = (const float*)d_in[13];
  const float* pec = (const float*)d_in[14];
  const float* diG = (const float*)d_in[15];
  const float* dic = (const float*)d_in[16];
  const float* dpG = (const float*)d_in[17];
  const float* dpc = (const float*)d_in[18];
  const float* rsG = (const float*)d_in[19];
  const float* rsc = (const float*)d_in[20];
  const float* rpG = (const float*)d_in[21];
  const float* rpc = (const float*)d_in[22];
  float* outp = (float*)d_out;

  // workspace allocator (needs ~205 MB of f32 scratch)
  float* base = (float*)d_ws;
  size_t pos = 0;
  auto alloc = [&](size_t n) { float* p = base + pos; pos += n; return p; };

  float* zerosBZ = alloc((size_t)NB * ZD);
  float* hA[5];
  for (int i = 0; i < 5; ++i) hA[i] = alloc((size_t)NB * HH);
  float* zh = alloc((size_t)NB * HH);
  float* q = alloc((size_t)NB * HH);
  float* x0 = alloc((size_t)NB * IMG);
  float* qa = alloc((size_t)NB * HH);
  float* a0pre = alloc((size_t)NB * AD);
  float* a0 = alloc((size_t)NB * AD);
  float* Ppe = alloc((size_t)NB * PE_T);
  float* Pdi = alloc((size_t)NB * DI_T);
  float* Pdp = alloc((size_t)NB * DP_T);
  float* Prs = alloc((size_t)NB * RN_T);
  float* Prp = alloc((size_t)NB * RN_T);
  float* erelu = alloc((size_t)NT * NB * HH);
  float* eout = alloc((size_t)NT * NB * HH);
  float* hs = alloc((size_t)NT * NB * ZD);
  float* hp = alloc((size_t)NT * NB * ZD);
  float* r1 = alloc((size_t)NT * NB * HH);
  float* s1 = alloc((size_t)NT * NB * HH);
  float* xt = alloc((size_t)NT * NB * IMG);
  float* apre = alloc((size_t)NT * NB * AD);
  float* at = alloc((size_t)NT * NB * AD);
  float* patch = alloc((size_t)NT * NB * IMG);
  float* orig = alloc((size_t)NT * NB * IMG);
  float* inp = alloc((size_t)NB * (IMG + AD));
  float* resid = alloc((size_t)NB * IMG);
  float* dPpe = alloc((size_t)NB * PE_T);
  float* dPdi = alloc((size_t)NB * DI_T);
  float* dPdp = alloc((size_t)NB * DP_T);
  float* dPrs = alloc((size_t)NB * RN_T);
  float* dPrp = alloc((size_t)NB * RN_T);
  float* dxt = alloc((size_t)NB * IMG);
  float* dat = alloc((size_t)NB * AD);
  float* dx_next = alloc((size_t)NB * IMG);
  float* da_next = alloc((size_t)NB * AD);
  float* dhs_carry = alloc((size_t)NB * ZD);
  float* dhp_carry = alloc((size_t)NB * ZD);
  float* dhs_tot = alloc((size_t)NB * ZD);
  float* dhp_tot = alloc((size_t)NB * ZD);
  float* bufS = alloc((size_t)NB * ZD);
  float* bufP = alloc((size_t)NB * ZD);
  float* bufE = alloc((size_t)NB * HH);
  float* tmpA = alloc((size_t)NB * (IMG + AD));
  float* tmpB = alloc((size_t)NB * ZD);
  float* du2 = alloc((size_t)NB * IMG);
  float* dv2 = alloc((size_t)NB * AD);
  float* dinp = alloc((size_t)NB * (IMG + AD));
  float* dzh = alloc((size_t)NB * HH);
  float* dh1 = alloc((size_t)NB * HH);
  float* dh2 = alloc((size_t)NB * HH);
  float* dzg = alloc((size_t)NB * ZD);
  float* zrun = alloc((size_t)NB * ZD);
  const int NZBIG = 64;
  float* part = alloc((size_t)NZBIG * NB * HH);

  auto MS = [&](float* p, size_t n) { hipMemsetAsync(p, 0, n * sizeof(float), stream); };
  auto gemm = [&](const float* A, const float* Bm, const float* bias, float* C,
                  int M, int N, int K, int act) {
    dim3 g((N + 31) / 32, (M + 15) / 16);
    nk_gemm<<<g, 32, 0, stream>>>(A, Bm, bias, C, M, N, K, act);
  };
  auto gemmbt = [&](const float* A, const float* G, float* Cout, int M, int N, int K,
                    int nz, int accOut) {
    long long kc = ((long long)K + nz - 1) / nz;
    kc = (kc + 3) & ~3LL;
    dim3 g((N + 15) / 16, (M + 15) / 16, nz);
    nk_gemm_bt<<<g, 32, 0, stream>>>(A, G, part, M, N, K, (int)kc);
    const int len = M * N;
    nk_reduce<<<(len + 255) / 256, 256, 0, stream>>>(part, nz, len, Cout, accOut);
  };
  auto bgemv = [&](const float* X, int xs, const float* P, long long pstr, long long woff,
                   long long boff, float* Y, int ys, int din, int dout, int acc, int act) {
    dim3 g((dout + 255) / 256, NB);
    nk_bgemv<<<g, 256, 0, stream>>>(X, xs, P, pstr, woff, boff, Y, ys, din, dout, acc, act);
  };
  auto bgemvT = [&](const float* DY, int ys, const float* P, long long pstr, long long woff,
                    float* DX, int xs, int din, int dout, int acc) {
    dim3 g((din + 255) / 256, NB);
    nk_bgemvT<<<g, 256, 0, stream>>>(DY, ys, P, pstr, woff, DX, xs, din, dout, acc);
  };
  auto bouter = [&](const float* X, int xs, const float* DY, int ys, float* dP,
                    long long pstr, long long woff, int din, int dout) {
    dim3 g((int)(((long long)din * dout + 255) / 256), NB);
    nk_bouter<<<g, 256, 0, stream>>>(X, xs, DY, ys, dP, pstr, woff, din, dout);
  };
  auto baddb = [&](const float* DY, int ys, float* dP, long long pstr, long long boff,
                   int dout) {
    dim3 g((dout + 255) / 256, NB);
    nk_baddb<<<g, 256, 0, stream>>>(DY, ys, dP, pstr, boff, dout);
  };
  auto EW = [&](int n) { return dim3((n + 255) / 256); };

  // ------------------------------------------------------------------ forward
  auto forward = [&](const float* zin, bool withOrig) {
    gemm(zin, hzW0, hzb0, hA[0], NB, HH, ZD, ACT_RELU);
    for (int i = 0; i < 4; ++i)
      gemm(hA[i], hzW + (size_t)i * HH * HH, hzb + (size_t)i * HH, hA[i + 1], NB, HH, HH,
           ACT_RELU);
    gemm(hA[4], hzW + (size_t)4 * HH * HH, hzb + (size_t)4 * HH, zh, NB, HH, HH, ACT_NONE);
    gemm(zh, ipW1, ipb1, q, NB, HH, HH, ACT_RELU);
    gemm(q, ipW2, ipb2, x0, NB, IMG, HH, ACT_SIG);
    gemm(zh, iolW1, iolb1, qa, NB, HH, HH, ACT_RELU);
    gemm(qa, iolW2, iolb2, a0pre, NB, AD, HH, ACT_NONE);
    nk_ew_sin<<<EW(NB * AD), 256, 0, stream>>>(a0, a0pre, NB * AD);
    gemm(zh, peG, pec, Ppe, NB, (int)PE_T, HH, ACT_NONE);
    gemm(zh, diG, dic, Pdi, NB, (int)DI_T, HH, ACT_NONE);
    gemm(zh, dpG, dpc, Pdp, NB, (int)DP_T, HH, ACT_NONE);
    gemm(zh, rsG, rsc, Prs, NB, (int)RN_T, HH, ACT_NONE);
    gemm(zh, rpG, rpc, Prp, NB, (int)RN_T, HH, ACT_NONE);
    for (int t = 0; t < NT; ++t) {
      const float* xprev = t ? xt + (size_t)(t - 1) * NB * IMG : x0;
      const float* aprev = t ? at + (size_t)(t - 1) * NB * AD : a0;
      float* erelu_t = erelu + (size_t)t * NB * HH;
      float* eout_t = eout + (size_t)t * NB * HH;
      float* hs_t = hs + (size_t)t * NB * ZD;
      float* hp_t = hp + (size_t)t * NB * ZD;
      float* r1_t = r1 + (size_t)t * NB * HH;
      float* s1_t = s1 + (size_t)t * NB * HH;
      float* xt_t = xt + (size_t)t * NB * IMG;
      float* apre_t = apre + (size_t)t * NB * AD;
      float* at_t = at + (size_t)t * NB * AD;
      const float* hsprev = t ? hs + (size_t)(t - 1) * NB * ZD : zerosBZ;
      const float* hpprev = t ? hp + (size_t)(t - 1) * NB * ZD : zerosBZ;
      nk_concat<<<EW(NB * (IMG + AD)), 256, 0, stream>>>(xprev, aprev, inp);
      bgemv(inp, IMG + AD, Ppe, PE_T, PE_W1, PE_B1, erelu_t, HH, IMG + AD, HH, 0, ACT_RELU);
      bgemv(erelu_t, HH, Ppe, PE_T, PE_W2, PE_B2, eout_t, HH, HH, HH, 0, ACT_NONE);
      bgemv(eout_t, HH, Prs, RN_T, RN_WI, RN_B, hs_t, ZD, HH, ZD, 0, ACT_NONE);
      bgemv(hsprev, ZD, Prs, RN_T, RN_WH, -1, hs_t, ZD, ZD, ZD, 1, ACT_TANH);
      bgemv(eout_t, HH, Prp, RN_T, RN_WI, RN_B, hp_t, ZD, HH, ZD, 0, ACT_NONE);
      bgemv(hpprev, ZD, Prp, RN_T, RN_WH, -1, hp_t, ZD, ZD, ZD, 1, ACT_TANH);
      bgemv(hs_t, ZD, Pdi, DI_T, DI_W1, DI_B1, r1_t, HH, ZD, HH, 0, ACT_RELU);
      bgemv(r1_t, HH, Pdi, DI_T, DI_W2, DI_B2, xt_t, IMG, HH, IMG, 0, ACT_SIG);
      bgemv(hp_t, ZD, Pdp, DP_T, DP_W1, DP_B1, s1_t, HH, ZD, HH, 0, ACT_RELU);
      bgemv(s1_t, HH, Pdp, DP_T, DP_W2, DP_B2, apre_t, AD, HH, AD, 0, ACT_NONE);
      nk_ew_sin<<<EW(NB * AD), 256, 0, stream>>>(at_t, apre_t, NB * AD);
      nk_patch_fwd<<<dim3(IMG / 256, NB), 256, 0, stream>>>(
          xt_t, at_t, patch + (size_t)t * NB * IMG);
    }
    if (withOrig)
      nk_zoom<<<dim3(IMG / 256, NB, NT), 256, 0, stream>>>(x, at, orig);
  };

  // ---------------------------------------------------------- phase A: z = 0
  MS(zerosBZ, (size_t)NB * ZD);
  forward(zerosBZ, false);
  nk_residual<<<EW(NB * IMG), 256, 0, stream>>>(patch, x, resid);

  // ------------------------------------------------------------ backward pass
  MS(dPpe, (size_t)NB * PE_T); MS(dPdi, (size_t)NB * DI_T); MS(dPdp, (size_t)NB * DP_T);
  MS(dPrs, (size_t)NB * RN_T); MS(dPrp, (size_t)NB * RN_T);
  MS(dx_next, (size_t)NB * IMG); MS(da_next, (size_t)NB * AD);
  MS(dhs_carry, (size_t)NB * ZD); MS(dhp_carry, (size_t)NB * ZD);

  for (int t = NT - 1; t >= 0; --t) {
    float* erelu_t = erelu + (size_t)t * NB * HH;
    float* eout_t = eout + (size_t)t * NB * HH;
    float* hs_t = hs + (size_t)t * NB * ZD;
    float* hp_t = hp + (size_t)t * NB * ZD;
    float* r1_t = r1 + (size_t)t * NB * HH;
    float* s1_t = s1 + (size_t)t * NB * HH;
    float* xt_t = xt + (size_t)t * NB * IMG;
    float* apre_t = apre + (size_t)t * NB * AD;
    float* at_t = at + (size_t)t * NB * AD;
    const float* hsprev = t ? hs + (size_t)(t - 1) * NB * ZD : zerosBZ;
    const float* hpprev = t ? hp + (size_t)(t - 1) * NB * ZD : zerosBZ;
    const float* xprev = t ? xt + (size_t)(t - 1) * NB * IMG : x0;
    const float* aprev = t ? at + (size_t)(t - 1) * NB * AD : a0;
    // sampler backward (dpatch_t == resid for all t)
    MS(dxt, (size_t)NB * IMG); MS(dat, (size_t)NB * AD);
    nk_patch_bwd<<<dim3(IMG / 256, NB), 256, 0, stream>>>(resid, xt_t, at_t, dxt, dat);
    nk_ew_add<<<EW(NB * IMG), 256, 0, stream>>>(dxt, dxt, dx_next, NB * IMG);
    nk_ew_add<<<EW(NB * AD), 256, 0, stream>>>(dat, dat, da_next, NB * AD);
    // x decoder (di)
    nk_ew_sigbwd<<<EW(NB * IMG), 256, 0, stream>>>(du2, xt_t, dxt, NB * IMG);
    bouter(r1_t, HH, du2, IMG, dPdi, DI_T, DI_W2, HH, IMG);
    baddb(du2, IMG, dPdi, DI_T, DI_B2, IMG);
    bgemvT(du2, IMG, Pdi, DI_T, DI_W2, tmpA, HH, HH, IMG, 0);
    nk_ew_relubwd<<<EW(NB * HH), 256, 0, stream>>>(tmpA, r1_t, tmpA, NB * HH);
    bouter(hs_t, ZD, tmpA, HH, dPdi, DI_T, DI_W1, ZD, HH);
    baddb(tmpA, HH, dPdi, DI_T, DI_B1, HH);
    bgemvT(tmpA, HH, Pdi, DI_T, DI_W1, tmpB, ZD, ZD, HH, 0);
    nk_ew_add<<<EW(NB * ZD), 256, 0, stream>>>(dhs_tot, tmpB, dhs_carry, NB * ZD);
    // a decoder (dp)
    nk_ew_sinbwd<<<EW(NB * AD), 256, 0, stream>>>(dv2, apre_t, dat, NB * AD);
    bouter(s1_t, HH, dv2, AD, dPdp, DP_T, DP_W2, HH, AD);
    baddb(dv2, AD, dPdp, DP_T, DP_B2, AD);
    bgemvT(dv2, AD, Pdp, DP_T, DP_W2, tmpA, HH, HH, AD, 0);
    nk_ew_relubwd<<<EW(NB * HH), 256, 0, stream>>>(tmpA, s1_t, tmpA, NB * HH);
    bouter(hp_t, ZD, tmpA, HH, dPdp, DP_T, DP_W1, ZD, HH);
    baddb(tmpA, HH, dPdp, DP_T, DP_B1, HH);
    bgemvT(tmpA, HH, Pdp, DP_T, DP_W1, tmpB, ZD, ZD, HH, 0);
    nk_ew_add<<<EW(NB * ZD), 256, 0, stream>>>(dhp_tot, tmpB, dhp_carry, NB * ZD);
    // RNN s
    nk_ew_tanhbwd<<<EW(NB * ZD), 256, 0, stream>>>(bufS, hs_t, dhs_tot, NB * ZD);
    bouter(eout_t, HH, bufS, ZD, dPrs, RN_T, RN_WI, HH, ZD);
    bouter(hsprev, ZD, bufS, ZD, dPrs, RN_T, RN_WH, ZD, ZD);
    baddb(bufS, ZD, dPrs, RN_T, RN_B, ZD);
    bgemvT(bufS, ZD, Prs, RN_T, RN_WI, bufE, HH, HH, ZD, 0);
    bgemvT(bufS, ZD, Prs, RN_T, RN_WH, dhs_carry, ZD, ZD, ZD, 0);
    // RNN p
    nk_ew_tanhbwd<<<EW(NB * ZD), 256, 0, stream>>>(bufP, hp_t, dhp_tot, NB * ZD);
    bouter(eout_t, HH, bufP, ZD, dPrp, RN_T, RN_WI, HH, ZD);
    bouter(hpprev, ZD, bufP, ZD, dPrp, RN_T, RN_WH, ZD, ZD);
    baddb(bufP, ZD, dPrp, RN_T, RN_B, ZD);
    bgemvT(bufP, ZD, Prp, RN_T, RN_WI, bufE, HH, HH, ZD, 1);  // de = de_s + de_p
    bgemvT(bufP, ZD, Prp, RN_T, RN_WH, dhp_carry, ZD, ZD, ZD, 0);
    // pe encoder
    bouter(erelu_t, HH, bufE, HH, dPpe, PE_T, PE_W2, HH, HH);
    baddb(bufE, HH, dPpe, PE_T, PE_B2, HH);
    bgemvT(bufE, HH, Ppe, PE_T, PE_W2, tmpA, HH, HH, HH, 0);
    nk_ew_relubwd<<<EW(NB * HH), 256, 0, stream>>>(tmpA, erelu_t, tmpA, NB * HH);
    nk_concat<<<EW(NB * (IMG + AD)), 256, 0, stream>>>(xprev, aprev, inp);
    bouter(inp, IMG + AD, tmpA, HH, dPpe, PE_T, PE_W1, IMG + AD, HH);
    baddb(tmpA, HH, dPpe, PE_T, PE_B1, HH);
    bgemvT(tmpA, HH, Ppe, PE_T, PE_W1, dinp, IMG + AD, IMG + AD, HH, 0);
    nk_split<<<EW(NB * (IMG + AD)), 256, 0, stream>>>(dinp, dx_next, da_next);
  }

  // dzh accumulation
  MS(dzh, (size_t)NB * HH);
  // x0 head
  nk_ew_sigbwd<<<EW(NB * IMG), 256, 0, stream>>>(du2, x0, dx_next, NB * IMG);
  gemmbt(du2, ipW2, dh1, NB, HH, IMG, 1, 0);
  nk_ew_relubwd<<<EW(NB * HH), 256, 0, stream>>>(dh1, q, dh1, NB * HH);
  gemmbt(dh1, ipW1, dzh, NB, HH, HH, 1, 1);
  // a0 head
  nk_ew_sinbwd<<<EW(NB * AD), 256, 0, stream>>>(dv2, a0pre, da_next, NB * AD);
  gemmbt(dv2, iolW2, dh1, NB, HH, AD, 1, 0);
  nk_ew_relubwd<<<EW(NB * HH), 256, 0, stream>>>(dh1, qa, dh1, NB * HH);
  gemmbt(dh1, iolW1, dzh, NB, HH, HH, 1, 1);
  // hypernetwork generators: dzh += dP @ G^T  (big split-K WMMA GEMMs)
  gemmbt(dPpe, peG, dzh, NB, HH, (int)PE_T, NZBIG, 1);
  gemmbt(dPdi, diG, dzh, NB, HH, (int)DI_T, NZBIG, 1);
  gemmbt(dPdp, dpG, dzh, NB, HH, (int)DP_T, NZBIG, 1);
  gemmbt(dPrs, rsG, dzh, NB, HH, (int)RN_T, NZBIG, 1);
  gemmbt(dPrp, rpG, dzh, NB, HH, (int)RN_T, NZBIG, 1);
  // head MLP backward
  gemmbt(dzh, hzW + (size_t)4 * HH * HH, dh1, NB, HH, HH, 1, 0);
  nk_ew_relubwd<<<EW(NB * HH), 256, 0, stream>>>(dh1, hA[4], dh1, NB * HH);
  for (int i = 3; i >= 0; --i) {
    gemmbt(dh1, hzW + (size_t)i * HH * HH, dh2, NB, HH, HH, 1, 0);
    nk_ew_relubwd<<<EW(NB * HH), 256, 0, stream>>>(dh2, hA[i], dh2, NB * HH);
    float* tmpSwap = dh1; dh1 = dh2; dh2 = tmpSwap;
  }
  gemmbt(dh1, hzW0, dzg, NB, ZD, HH, 1, 0);
  nk_ew_neg<<<EW(NB * ZD), 256, 0, stream>>>(zrun, dzg, NB * ZD);

  // ------------------------------------------------------- phase C: z = -grad
  forward(zrun, true);
  nk_pack<<<(OUT_TOT + 255) / 256, 256, 0, stream>>>(hs, at, xt, patch, orig, outp);
}